// AirNetworkSystem_29841432772826
// MI455X (gfx1250) — compile-verified
//
#include <hip/hip_runtime.h>
#include <math.h>

// ---------------------------------------------------------------------------
// AirNetworkSystem equilibrium solver, MI455X (gfx1250, wave32).
//
// residual(f) = P*(s^2 - (f/Q)^2) - [(Rduct_s + Rdmp_s(pos))*f^2 + Rlin_s*f]
//                                 - [(Rduct_e + Rdmp_e(pos))*f^2 + Rlin_e*f]
// Damper resistance depends only on damper position -> precompute
// sumR = Rduct + Ropen*exp(k*(1-pos)) once per element (same op order as ref).
//
// Global coupling: done/stall gating uses max(|residual|) over ALL elements.
// Each k_iterate applies the PREVIOUS iteration's gated update, computes the
// new residual, and reduces the global max; the LAST workgroup to arrive
// (atomic counter + agent-scope fences) runs the scalar done/stall/alpha
// state machine in-kernel, eliminating 50 one-thread launches from the chain.
// Working set ~40 MB -> fully L2-resident (192 MB) across the 50 iterations.
// ---------------------------------------------------------------------------

#define FP_TOL       1e-3f
#define FP_STALL_EPS 1e-6f
#define N_ITER       50

struct ScalarState {
  unsigned max_bits;   // running max of |residual| (nonneg float as ordered uint)
  unsigned arrived;    // workgroup arrival counter for the last-block step
  float    prev_err;
  int      done;
  int      stall;
  float    alpha;      // step size for the pending (gated) update
  int      do_update;  // 1 = apply pending update, 0 = frozen
};

struct Consts {
  float Rlin_s, Rlin_e;   // |p1| per branch
  float P, Q;             // design pressure, design flow
  float flow_scale;       // Q / (P + 1e-6)
  float clip_lo, clip_hi; // 0.01, 1.5*Q
};

// ---------------------------------------------------------------- init ------
__global__ void k_init(const float* __restrict__ sp, const float* __restrict__ ep,
                       const float* __restrict__ fp,
                       ScalarState* __restrict__ st, Consts* __restrict__ c) {
  st->max_bits  = 0u;
  st->arrived   = 0u;
  st->prev_err  = __int_as_float(0x7f800000); // +inf
  st->done      = 0;
  st->stall     = 0;
  st->alpha     = 0.0f;
  st->do_update = 0;                          // iteration 0 applies no update
  const float Q = fp[0], P = fp[1];
  c->Rlin_s     = fabsf(sp[1]);
  c->Rlin_e     = fabsf(ep[1]);
  c->P          = P;
  c->Q          = Q;
  c->flow_scale = Q / (P + 1e-6f);
  c->clip_lo    = 0.01f;
  c->clip_hi    = Q * 1.5f;
}

// ---------------------------------------------------------- precompute ------
// flow_init = speed * Q ; sumR = exp(p0) + exp(p2)*exp(p3*(1-pos)) per branch.
__global__ void __launch_bounds__(256)
k_precompute(const float* __restrict__ fan_speed,
             const float* __restrict__ spos, const float* __restrict__ epos,
             const float* __restrict__ sp,   const float* __restrict__ ep,
             const float* __restrict__ fp,
             float* __restrict__ flow,
             float* __restrict__ sumRs, float* __restrict__ sumRe, int n) {
  const int i    = blockIdx.x * blockDim.x + threadIdx.x;
  const int base = i * 4;
  if (base >= n) return;

  const float Rduct_s = expf(sp[0]), Ropen_s = expf(sp[2]), ks = sp[3];
  const float Rduct_e = expf(ep[0]), Ropen_e = expf(ep[2]), ke = ep[3];
  const float Q = fp[0];

  if (base + 3 < n) {  // 128-bit path (global_load_b128 / global_store_b128)
    const float4 spd = ((const float4*)fan_speed)[i];
    const float4 ps  = ((const float4*)spos)[i];
    const float4 pe  = ((const float4*)epos)[i];
    float s[4]  = {spd.x, spd.y, spd.z, spd.w};
    float a[4]  = {ps.x, ps.y, ps.z, ps.w};
    float b[4]  = {pe.x, pe.y, pe.z, pe.w};
    float f[4], rs[4], re[4];
#pragma unroll
    for (int j = 0; j < 4; ++j) {
      f[j]  = s[j] * Q;
      rs[j] = Rduct_s + Ropen_s * expf(ks * (1.0f - a[j]));
      re[j] = Rduct_e + Ropen_e * expf(ke * (1.0f - b[j]));
    }
    ((float4*)flow )[i] = make_float4(f[0],  f[1],  f[2],  f[3]);
    ((float4*)sumRs)[i] = make_float4(rs[0], rs[1], rs[2], rs[3]);
    ((float4*)sumRe)[i] = make_float4(re[0], re[1], re[2], re[3]);
  } else {             // scalar tail
    for (int k = base; k < n; ++k) {
      flow[k]  = fan_speed[k] * Q;
      sumRs[k] = Rduct_s + Ropen_s * expf(ks * (1.0f - spos[k]));
      sumRe[k] = Rduct_e + Ropen_e * expf(ke * (1.0f - epos[k]));
    }
  }
}

// ------------------------------------------------------------- iterate ------
__device__ __forceinline__ void elem_step(float& f, float& r, float s,
                                          float Cs, float Ce,
                                          float alpha, int upd,
                                          float P, float Q, float fsc,
                                          float lo, float hi,
                                          float Rls, float Rle, float& m) {
  if (upd) {  // apply previous iteration's gated update (ref op order)
    float fu = f + (alpha * r) * fsc;
    f = fminf(fmaxf(fu, lo), hi);
  }
  const float q    = f / Q;
  const float fanp = P * (s * s - q * q);
  const float af   = fabsf(f);
  const float dps  = (Cs * af) * af + Rls * af;
  const float dpe  = (Ce * af) * af + Rle * af;
  r = fanp - dps - dpe;
  m = fmaxf(m, fabsf(r));
}

__global__ void __launch_bounds__(256)
k_iterate(const float* __restrict__ fan_speed,
          float* __restrict__ flow, float* __restrict__ resid,
          const float* __restrict__ sumRs, const float* __restrict__ sumRe,
          const Consts* __restrict__ c, ScalarState* __restrict__ st,
          int n, int iter) {
  const int i    = blockIdx.x * blockDim.x + threadIdx.x;
  const int base = i * 4;
  float m = 0.0f;

  if (base < n) {
    const float alpha = st->alpha;     // uniform scalars (L2-hot)
    const int   upd   = st->do_update;
    const float P = c->P, Q = c->Q, fsc = c->flow_scale;
    const float lo = c->clip_lo, hi = c->clip_hi;
    const float Rls = c->Rlin_s, Rle = c->Rlin_e;

    if (base + 3 < n) {
      float4 f4 = ((float4*)flow )[i];
      float4 r4 = ((float4*)resid)[i];
      const float4 s4 = ((const float4*)fan_speed)[i];
      const float4 a4 = ((const float4*)sumRs)[i];
      const float4 b4 = ((const float4*)sumRe)[i];
      float f[4]  = {f4.x, f4.y, f4.z, f4.w};
      float r[4]  = {r4.x, r4.y, r4.z, r4.w};
      float s[4]  = {s4.x, s4.y, s4.z, s4.w};
      float Cs[4] = {a4.x, a4.y, a4.z, a4.w};
      float Ce[4] = {b4.x, b4.y, b4.z, b4.w};
#pragma unroll
      for (int j = 0; j < 4; ++j)
        elem_step(f[j], r[j], s[j], Cs[j], Ce[j], alpha, upd,
                  P, Q, fsc, lo, hi, Rls, Rle, m);
      ((float4*)flow )[i] = make_float4(f[0], f[1], f[2], f[3]);
      ((float4*)resid)[i] = make_float4(r[0], r[1], r[2], r[3]);
    } else {
      for (int k = base; k < n; ++k) {
        float f = flow[k], r = resid[k];
        elem_step(f, r, fan_speed[k], sumRs[k], sumRe[k], alpha, upd,
                  P, Q, fsc, lo, hi, Rls, Rle, m);
        flow[k] = f; resid[k] = r;
      }
    }
  }

  // wave32 max-reduction tree (|r| >= 0 -> bit pattern is order-preserving)
  unsigned mb = __float_as_uint(m);
#pragma unroll
  for (int off = 16; off > 0; off >>= 1) {
    unsigned o = __shfl_xor(mb, off, 32);
    mb = (o > mb) ? o : mb;
  }
  __shared__ unsigned wmax[8];           // 256 threads = 8 waves on CDNA5
  const int wid = threadIdx.x >> 5, lid = threadIdx.x & 31;
  if (lid == 0) wmax[wid] = mb;
  __syncthreads();
  if (wid == 0) {
    unsigned v = (lid < 8) ? wmax[lid] : 0u;
#pragma unroll
    for (int off = 4; off > 0; off >>= 1) {
      unsigned o = __shfl_xor(v, off, 32);
      v = (o > v) ? o : v;
    }
    if (lid == 0) {
      if (v) atomicMax(&st->max_bits, v);
      __threadfence();                   // publish this block's max (agent scope)
      const unsigned prev = __hip_atomic_fetch_add(
          &st->arrived, 1u, __ATOMIC_ACQ_REL, __HIP_MEMORY_SCOPE_AGENT);
      if (prev == (unsigned)(gridDim.x - 1)) {
        // ---- last workgroup: scalar done/stall/alpha state machine ----
        __threadfence();
        const float max_err = __uint_as_float(__hip_atomic_load(
            &st->max_bits, __ATOMIC_RELAXED, __HIP_MEMORY_SCOPE_AGENT));
        const float prv   = st->prev_err;
        const int   done  = st->done;
        const int   stall = st->stall;

        const int done_tol  = done || (max_err < FP_TOL);
        const int stall_new = done_tol ? stall
                              : ((fabsf(max_err - prv) < FP_STALL_EPS) ? stall + 1 : 0);
        const int done_new  = done_tol || (stall_new > 10);

        st->prev_err  = done_new ? prv : max_err;
        st->done      = done_new;
        st->stall     = stall_new;
        st->alpha     = fmaxf(0.5f * powf(0.95f, (float)iter), 0.05f);
        st->do_update = done_new ? 0 : 1;
        st->max_bits  = 0u;              // re-arm accumulator
        st->arrived   = 0u;              // re-arm arrival counter
      }
    }
  }
}

// ------------------------------------------------------------ finalize ------
__global__ void __launch_bounds__(256)
k_finalize(float* __restrict__ flow, const float* __restrict__ resid,
           const Consts* __restrict__ c, const ScalarState* __restrict__ st,
           int n) {
  const int i    = blockIdx.x * blockDim.x + threadIdx.x;
  const int base = i * 4;
  if (base >= n) return;
  const float alpha = st->alpha;
  const int   upd   = st->do_update;
  const float fsc = c->flow_scale, lo = c->clip_lo, hi = c->clip_hi;

  if (base + 3 < n) {
    float4 f4 = ((float4*)flow)[i];
    const float4 r4 = ((const float4*)resid)[i];
    float f[4] = {f4.x, f4.y, f4.z, f4.w};
    float r[4] = {r4.x, r4.y, r4.z, r4.w};
#pragma unroll
    for (int j = 0; j < 4; ++j)
      if (upd) f[j] = fminf(fmaxf(f[j] + (alpha * r[j]) * fsc, lo), hi);
    ((float4*)flow)[i] = make_float4(f[0], f[1], f[2], f[3]);
  } else {
    for (int k = base; k < n; ++k)
      if (upd) flow[k] = fminf(fmaxf(flow[k] + (alpha * resid[k]) * fsc, lo), hi);
  }
}

// -------------------------------------------------------------- launch ------
extern "C" void kernel_launch(void* const* d_in, const int* in_sizes, int n_in,
                              void* d_out, int out_size, void* d_ws, size_t ws_size,
                              hipStream_t stream) {
  const float* fan_speed = (const float*)d_in[0];
  const float* spos      = (const float*)d_in[1];
  const float* epos      = (const float*)d_in[2];
  const float* sp        = (const float*)d_in[3];
  const float* ep        = (const float*)d_in[4];
  const float* fp        = (const float*)d_in[5];
  float*       flow      = (float*)d_out;          // flow state lives in d_out
  const int    n         = in_sizes[0];

  char* ws = (char*)d_ws;
  ScalarState* st = (ScalarState*)ws;
  Consts*      c  = (Consts*)(ws + 64);
  const size_t arrB = ((size_t)n * sizeof(float) + 255) & ~(size_t)255;
  float* resid = (float*)(ws + 256);
  float* sumRs = (float*)(ws + 256 + arrB);
  float* sumRe = (float*)(ws + 256 + 2 * arrB);

  const int nvec   = (n + 3) / 4;
  const int blocks = (nvec + 255) / 256;

  k_init<<<1, 1, 0, stream>>>(sp, ep, fp, st, c);
  k_precompute<<<blocks, 256, 0, stream>>>(fan_speed, spos, epos, sp, ep, fp,
                                           flow, sumRs, sumRe, n);
  for (int i = 0; i < N_ITER; ++i) {
    k_iterate<<<blocks, 256, 0, stream>>>(fan_speed, flow, resid,
                                          sumRs, sumRe, c, st, n, i);
  }
  k_finalize<<<blocks, 256, 0, stream>>>(flow, resid, c, st, n);
}